// SpikeFP32Softmax_57526791962936
// MI455X (gfx1250) — compile-verified
//
#include <hip/hip_runtime.h>

typedef float v2f __attribute__((ext_vector_type(2)));
typedef float v8f __attribute__((ext_vector_type(8)));

#define RDIM 1024          // rows (softmax/reduction dim)
#define NDIM 1024          // columns
#define COLS 16            // columns handled per block (== WMMA N)
#define THREADS 256        // 8 waves (wave32)
#define RLANES (THREADS / COLS)            // 16 row-lanes
#define RPT (RDIM / RLANES)                // 64 rows per thread

// Decode 32 MSB-first 0/1 float pulses (128B) into the packed IEEE754 word.
// Pulses are exactly 0.0f / 1.0f, so bit = (asuint >> 23) & 1 (exponent LSB).
__device__ __forceinline__ unsigned pack_bits(const float4* __restrict__ p) {
  unsigned u = 0;
#pragma unroll
  for (int j = 0; j < 8; ++j) {
    float4 v = p[j];
    u = (u << 1) | ((__float_as_uint(v.x) >> 23) & 1u);
    u = (u << 1) | ((__float_as_uint(v.y) >> 23) & 1u);
    u = (u << 1) | ((__float_as_uint(v.z) >> 23) & 1u);
    u = (u << 1) | ((__float_as_uint(v.w) >> 23) & 1u);
  }
  return u;
}

// Encode packed word back to 32 MSB-first 0/1 float pulses (8x b128 stores).
__device__ __forceinline__ void store_bits(float4* __restrict__ p, unsigned u) {
#pragma unroll
  for (int j = 0; j < 8; ++j) {
    const int k = 4 * j;  // bit index of .x (0 == MSB)
    float4 o;
    o.x = (float)((u >> (31 - k)) & 1u);
    o.y = (float)((u >> (30 - k)) & 1u);
    o.z = (float)((u >> (29 - k)) & 1u);
    o.w = (float)((u >> (28 - k)) & 1u);
    p[j] = o;
  }
}

extern "C" __global__ __launch_bounds__(THREADS)
void spikefp32_softmax_fused(const float* __restrict__ x,
                             float* __restrict__ out) {
  // Dynamic LDS: exp tile (1024x16 f32 = 64KB) + 256 partials + 16 recips.
  extern __shared__ float smem[];
  float* exp_tile = smem;                       // [RDIM][COLS]
  float* part     = smem + RDIM * COLS;         // [RLANES][COLS]
  float* rsum     = part + THREADS;             // [COLS]

  const int tid   = threadIdx.x;
  const int c     = tid & (COLS - 1);           // column within tile
  const int rl    = tid >> 4;                   // row-lane 0..15
  const int b     = blockIdx.x / (NDIM / COLS);
  const int nbase = (blockIdx.x % (NDIM / COLS)) * COLS;

  const float4* __restrict__ x4 = (const float4*)x;
  float4* __restrict__ o4 = (float4*)out;

  // ---- Phase 1: decode -> exp -> LDS, per-thread partial sums ----
  float psum = 0.0f;
#pragma unroll 4
  for (int i = 0; i < RPT; ++i) {
    const int  r = rl + i * RLANES;
    const long e = (long)(b * RDIM + r) * NDIM + (nbase + c);
    const float4* p = x4 + e * 8;
    if (i + 1 < RPT)  // next read is +2MB: hardware won't prefetch that stride
      __builtin_prefetch((const void*)(p + (long)RLANES * NDIM * 8), 0, 1);
    const unsigned u = pack_bits(p);
    const float ex = __expf(__uint_as_float(u));
    exp_tile[r * COLS + c] = ex;
    psum += ex;
  }
  part[rl * COLS + c] = psum;
  __syncthreads();

  // ---- Phase 2: 16x16 partial tile column-sum via V_WMMA_F32_16X16X4_F32.
  // D = A(16x4 ones in row M=0) x B(4x16 partials) + C, accumulated 4x (K=16).
  // A row M=0 lives in lanes {0,16}; setting ALL K slots of that row to 1.0
  // makes the sum invariant to the B K-slot lane mapping. D[0][n] -> VGPR0,
  // lane n (n=0..15). Whole-wave condition => EXEC all-ones inside (WMMA req).
  if (tid < 32) {
    const int lane = tid;
    const int col  = lane & 15;
    const float av = (col == 0) ? 1.0f : 0.0f;
    v2f A; A.x = av; A.y = av;
    v8f C = {0.f, 0.f, 0.f, 0.f, 0.f, 0.f, 0.f, 0.f};
#pragma unroll
    for (int kk = 0; kk < 4; ++kk) {
      const int rbase = kk * 4 + ((lane >> 4) << 1);  // lanes<16: +0,1; >=16: +2,3
      v2f B;
      B.x = part[(rbase + 0) * COLS + col];
      B.y = part[(rbase + 1) * COLS + col];
      C = __builtin_amdgcn_wmma_f32_16x16x4_f32(
          /*neg_a=*/false, A, /*neg_b=*/false, B,
          /*c_mod=*/(short)0, C, /*reuse_a=*/false, /*reuse_b=*/false);
    }
    if (lane < 16) rsum[lane] = 1.0f / C[0];
  }
  __syncthreads();

  // ---- Phase 3: divide + re-encode + store ----
  const float rs = rsum[c];
#pragma unroll 4
  for (int i = 0; i < RPT; ++i) {
    const int  r = rl + i * RLANES;
    const long e = (long)(b * RDIM + r) * NDIM + (nbase + c);
    const unsigned u = __float_as_uint(exp_tile[r * COLS + c] * rs);
    store_bits(o4 + e * 8, u);
  }
}

extern "C" void kernel_launch(void* const* d_in, const int* in_sizes, int n_in,
                              void* d_out, int out_size, void* d_ws, size_t ws_size,
                              hipStream_t stream) {
  (void)n_in; (void)out_size; (void)d_ws; (void)ws_size;
  const float* x = (const float*)d_in[0];
  float* out = (float*)d_out;

  const int total = in_sizes[0];                 // B * 1024 * 1024 * 32
  const int B = total / (RDIM * NDIM * 32);      // == 4 for the reference

  const dim3 grid(B * (NDIM / COLS));            // 256 blocks
  const size_t lds_bytes =
      (size_t)(RDIM * COLS + THREADS + COLS) * sizeof(float);  // 66,624 B (<320KB WGP LDS)

  spikefp32_softmax_fused<<<grid, THREADS, lds_bytes, stream>>>(x, out);
}